// KalmanFilter_23888608101237
// MI455X (gfx1250) — compile-verified
//
#include <hip/hip_runtime.h>

// Kalman filter on gfx1250 (MI455X).
// One block per batch element (128 blocks, 512 threads = 16 wave32).
// All 64x64 matmuls run on V_WMMA_F32_16X16X4_F32: each wave owns one 16x16
// output tile and accumulates over K=64 with 16 chained WMMAs.
// u_t / y_t staged into LDS with global_load_async_to_lds_b32 (ASYNCcnt path).

typedef __attribute__((ext_vector_type(2))) float v2f;
typedef __attribute__((ext_vector_type(8))) float v8f;

#define DD       64
#define SEQLEN   256
#define NTHREADS 512   // 16 waves

// Store modes for the WMMA tile writeback (fuses elementwise epilogues).
#define STORE_PLAIN  0
#define STORE_ADD    1   // C = acc + Add
#define STORE_IMINUS 2   // C = I - acc   (for IKH = I - K@H)

// ---------------------------------------------------------------------------
// WMMA fragment gather from a 64x64 row-major LDS matrix.
// A-matrix 16x4 f32 layout: lanes 0-15 -> K={kk,kk+1}, lanes 16-31 -> K={kk+2,kk+3}
// B-matrix 4x16 f32 layout: VGPR v lanes 0-15 -> K=kk+v, lanes 16-31 -> K=kk+2+v
// C/D 16x16 f32 layout:     VGPR r lanes 0-15 -> M=r, lanes 16-31 -> M=r+8
// TA/TB select transposed indexing so we never materialize transposes.
// ---------------------------------------------------------------------------
template<bool TA, bool TB>
__device__ inline v8f tile_mm_acc(const float* __restrict__ A,
                                  const float* __restrict__ B,
                                  int ti, int tj, v8f acc) {
  const int lane = threadIdx.x & 31;
  const int half = lane >> 4;          // 0 or 1
  const int idx  = lane & 15;
  const int row  = ti * 16 + idx;      // M index for A fragment
  const int col  = tj * 16 + idx;      // N index for B fragment
#pragma unroll
  for (int kk = 0; kk < DD; kk += 4) {
    const int k0 = kk + 2 * half;
    v2f a, b;
    if (TA) { a.x = A[(k0    ) * DD + row]; a.y = A[(k0 + 1) * DD + row]; }
    else    { a.x = A[row * DD + (k0    )]; a.y = A[row * DD + (k0 + 1)]; }
    if (TB) { b.x = B[col * DD + (k0    )]; b.y = B[col * DD + (k0 + 1)]; }
    else    { b.x = B[(k0    ) * DD + col]; b.y = B[(k0 + 1) * DD + col]; }
    // 8 args: (neg_a, A, neg_b, B, c_mod, C, reuse_a, reuse_b)
    acc = __builtin_amdgcn_wmma_f32_16x16x4_f32(
        false, a, false, b, (short)0, acc, false, false);
  }
  return acc;
}

template<int MODE>
__device__ inline void tile_store(float* __restrict__ C, int ti, int tj,
                                  v8f acc, const float* __restrict__ Add) {
  const int lane = threadIdx.x & 31;
  const int half = lane >> 4;
  const int n    = tj * 16 + (lane & 15);
#pragma unroll
  for (int r = 0; r < 8; ++r) {
    const int m = ti * 16 + r + 8 * half;
    float v = acc[r];
    if (MODE == STORE_ADD)    v += Add[m * DD + n];
    if (MODE == STORE_IMINUS) v = ((m == n) ? 1.0f : 0.0f) - v;
    C[m * DD + n] = v;
  }
}

// Block-level 64x64 GEMM: C = epilogue(op(A) * op(B)). 16 waves, one tile each.
template<bool TA, bool TB, int MODE>
__device__ inline void mm64(float* __restrict__ C,
                            const float* __restrict__ A,
                            const float* __restrict__ B,
                            const float* __restrict__ Add) {
  __syncthreads();                      // inputs ready for all waves
  const int wave = threadIdx.x >> 5;    // 0..15
  const int ti = wave >> 2;
  const int tj = wave & 3;
  v8f acc;
#pragma unroll
  for (int r = 0; r < 8; ++r) acc[r] = 0.0f;
  acc = tile_mm_acc<TA, TB>(A, B, ti, tj, acc);
  tile_store<MODE>(C, ti, tj, acc, Add);
  __syncthreads();                      // output visible to all waves
}

// ---------------------------------------------------------------------------
__global__ __launch_bounds__(NTHREADS)
void kalman_wmma_kernel(const float* __restrict__ x0,
                        const float* __restrict__ U,
                        const float* __restrict__ Y,
                        const float* __restrict__ Fg,
                        const float* __restrict__ Bg,
                        const float* __restrict__ Hg,
                        const float* __restrict__ LQg,
                        const float* __restrict__ LRg,
                        float* __restrict__ preds,
                        float* __restrict__ covs) {
  extern __shared__ float smem[];
  float* cF    = smem;                // 4096 floats each
  float* cB    = cF  + 4096;
  float* cH    = cB  + 4096;
  float* cQ    = cH  + 4096;
  float* cR    = cQ  + 4096;
  float* P     = cR  + 4096;
  float* Pp    = P   + 4096;          // P_pred
  float* T     = Pp  + 4096;          // temp / IKH
  float* T2    = T   + 4096;          // temp / S / S_inv
  float* Km    = T2  + 4096;          // Kalman gain
  float* Saug  = Km  + 4096;          // 64 x 128 Gauss-Jordan workspace
  float* fac   = Saug + 64 * 128;     // 64 elimination factors (pre-scaled)
  float* rowk  = fac + 64;            // 128 staged pivot-row entries
  float* xprev = rowk + 128;
  float* xpred = xprev + 64;
  float* innov = xpred + 64;
  float* xnew  = innov + 64;
  float* uvec  = xnew + 64;
  float* yvec  = uvec + 64;

  const int b   = blockIdx.x;
  const int tid = threadIdx.x;

  // ---- one-time setup: params to LDS, Q = LQ LQ^T, R = LR LR^T, P = I ----
  for (int i = tid; i < DD * DD; i += NTHREADS) {
    cF[i] = Fg[i];
    cB[i] = Bg[i];
    cH[i] = Hg[i];
    T[i]  = LQg[i];
    T2[i] = LRg[i];
    P[i]  = ((i >> 6) == (i & 63)) ? 1.0f : 0.0f;
  }
  if (tid < DD) xprev[tid] = x0[(size_t)b * DD + tid];
  mm64<false, true, STORE_PLAIN>(cQ, T,  T,  nullptr);   // Q = LQ @ LQ^T
  mm64<false, true, STORE_PLAIN>(cR, T2, T2, nullptr);   // R = LR @ LR^T

  const float* Ub   = U     + (size_t)b * SEQLEN * DD;
  const float* Yb   = Y     + (size_t)b * SEQLEN * DD;
  float* predb      = preds + (size_t)b * SEQLEN * DD;
  float* covb       = covs  + (size_t)b * SEQLEN * DD * DD;

  for (int t = 0; t < SEQLEN; ++t) {
    // ---- stage u_t, y_t via async global->LDS (ASYNCcnt-tracked) ----
    if (tid < DD) {
      const unsigned lu = (unsigned)(size_t)&uvec[tid];
      const unsigned ly = (unsigned)(size_t)&yvec[tid];
      const float* gu = &Ub[t * DD + tid];
      const float* gy = &Yb[t * DD + tid];
      asm volatile("global_load_async_to_lds_b32 %0, %1, off"
                   :: "v"(lu), "v"(gu) : "memory");
      asm volatile("global_load_async_to_lds_b32 %0, %1, off"
                   :: "v"(ly), "v"(gy) : "memory");
      asm volatile("s_wait_asynccnt 0" ::: "memory");
      if (t + 1 < SEQLEN) {   // warm L2/WGP$ for next step
        __builtin_prefetch(&Ub[(t + 1) * DD + tid], 0, 1);
        __builtin_prefetch(&Yb[(t + 1) * DD + tid], 0, 1);
      }
    }
    __syncthreads();

    // x_pred = x_prev @ F^T + u @ B^T
    if (tid < DD) {
      float s = 0.0f;
      for (int j = 0; j < DD; ++j)
        s += xprev[j] * cF[tid * DD + j] + uvec[j] * cB[tid * DD + j];
      xpred[tid] = s;
    }

    // P_pred = F P F^T + Q
    mm64<false, false, STORE_PLAIN>(T,  cF, P,  nullptr);
    mm64<false, true,  STORE_ADD  >(Pp, T,  cF, cQ);
    // S = H P_pred H^T + R   (into T2)
    mm64<false, false, STORE_PLAIN>(T,  cH, Pp, nullptr);
    mm64<false, true,  STORE_ADD  >(T2, T,  cH, cR);

    // ---- S_inv via Gauss-Jordan on [S | I] (S is SPD: diagonal pivots) ----
    for (int i = tid; i < DD * 128; i += NTHREADS) {
      const int r = i >> 7, c = i & 127;
      Saug[i] = (c < DD) ? T2[r * DD + c] : ((c - DD == r) ? 1.0f : 0.0f);
    }
    __syncthreads();
    for (int k = 0; k < DD; ++k) {
      // ip valid: previous iteration (or init) ended with a barrier.
      const float ip = 1.0f / Saug[k * 128 + k];
      if (tid < 128)      rowk[tid] = Saug[k * 128 + tid];
      else if (tid < 192) fac[tid - 128] = Saug[(tid - 128) * 128 + k] * ip;
      __syncthreads();
      // Single race-free pass: normalize row k, rank-1 update all other rows.
      for (int i = tid; i < DD * 128; i += NTHREADS) {
        const int r = i >> 7, c = i & 127;
        Saug[i] = (r == k) ? rowk[c] * ip : Saug[i] - fac[r] * rowk[c];
      }
      __syncthreads();
    }
    // compact S_inv into T2
    for (int i = tid; i < DD * DD; i += NTHREADS)
      T2[i] = Saug[(i >> 6) * 128 + DD + (i & 63)];

    // T = H^T @ S_inv ; K = P_pred @ T
    mm64<true,  false, STORE_PLAIN>(T,  cH, T2, nullptr);
    mm64<false, false, STORE_PLAIN>(Km, Pp, T,  nullptr);

    // innov = y - x_pred @ H^T ; x_new = x_pred + innov @ K
    if (tid < DD) {
      float s = 0.0f;
      for (int j = 0; j < DD; ++j) s += xpred[j] * cH[tid * DD + j];
      innov[tid] = yvec[tid] - s;
    }
    __syncthreads();
    if (tid < DD) {
      float s = xpred[tid];
      for (int j = 0; j < DD; ++j) s += innov[j] * Km[j * DD + tid];
      xnew[tid] = s;
    }

    // IKH = I - K @ H   (epilogue fused into WMMA store)
    mm64<false, false, STORE_IMINUS>(T, Km, cH, nullptr);
    // P_new = IKH P_pred IKH^T + K R K^T   (Joseph form)
    mm64<false, false, STORE_PLAIN>(T2, T,  Pp, nullptr);   // T2 = IKH @ P_pred
    mm64<false, true,  STORE_PLAIN>(Pp, T2, T,  nullptr);   // Pp = T2 @ IKH^T
    mm64<false, false, STORE_PLAIN>(T2, Km, cR, nullptr);   // T2 = K @ R
    mm64<false, true,  STORE_ADD  >(Pp, T2, Km, Pp);        // Pp += T2 @ K^T
    // symmetrize into P
    for (int i = tid; i < DD * DD; i += NTHREADS) {
      const int r = i >> 6, c = i & 63;
      P[i] = 0.5f * (Pp[i] + Pp[c * DD + r]);
    }
    __syncthreads();

    // ---- emit outputs; carry state ----
    if (tid < DD) {
      predb[t * DD + tid] = xnew[tid];
      xprev[tid] = xnew[tid];
    }
    {
      float* dst = covb + (size_t)t * DD * DD;
      for (int i = tid * 4; i < DD * DD; i += NTHREADS * 4) {
        const float4 v = *reinterpret_cast<const float4*>(&P[i]);
        *reinterpret_cast<float4*>(&dst[i]) = v;   // b128 coalesced stores
      }
    }
    __syncthreads();
  }
}

extern "C" void kernel_launch(void* const* d_in, const int* in_sizes, int n_in,
                              void* d_out, int out_size, void* d_ws, size_t ws_size,
                              hipStream_t stream) {
  const float* x0 = (const float*)d_in[0];
  const float* U  = (const float*)d_in[1];
  const float* Y  = (const float*)d_in[2];
  const float* F  = (const float*)d_in[3];
  const float* Bm = (const float*)d_in[4];
  const float* H  = (const float*)d_in[5];
  const float* LQ = (const float*)d_in[6];
  const float* LR = (const float*)d_in[7];

  float* preds = (float*)d_out;                       // [128,256,64]
  float* covs  = preds + (size_t)128 * 256 * 64;      // [128,256,64,64]

  // 10 full 64x64 mats + 64x128 aug + staging + vectors  (~195 KB of LDS,
  // fits the 320 KB/WGP budget; one workgroup per WGP)
  const size_t shmem =
      (size_t)(10 * 4096 + 64 * 128 + 64 + 128 + 6 * 64) * sizeof(float);

  hipLaunchKernelGGL(kalman_wmma_kernel, dim3(128), dim3(NTHREADS), shmem,
                     stream, x0, U, Y, F, Bm, H, LQ, LR, preds, covs);
}